// ImportanceLearning_47124381172385
// MI455X (gfx1250) — compile-verified
//
#include <hip/hip_runtime.h>

typedef __attribute__((ext_vector_type(2))) float v2f;
typedef __attribute__((ext_vector_type(4))) float v4f;
typedef __attribute__((ext_vector_type(8))) float v8f;

#define FTN 8
#define SEG 128
#define DCOLS 1024
#define ROWS_PER_WAVE 16
#define WAVES_PER_BLOCK 8
#define ROWS_PER_BLOCK (ROWS_PER_WAVE * WAVES_PER_BLOCK)

__global__ __launch_bounds__(256) void importance_gate_kernel(
    const float* __restrict__ x,
    const float* __restrict__ W,     // [FTN, FTN] row-major: W[out][in]
    const float* __restrict__ bias,  // [FTN]
    float* __restrict__ out)
{
    __shared__ float smax[ROWS_PER_BLOCK][FTN];    // segment maxima
    __shared__ float sscale[ROWS_PER_BLOCK][FTN];  // sigmoid gate values

    const int lane = threadIdx.x & 31;
    const int wave = threadIdx.x >> 5;
    const long long rowBase = (long long)blockIdx.x * ROWS_PER_BLOCK
                            + (long long)wave * ROWS_PER_WAVE;

    // ---------------- Phase 1: per-segment max ------------------------------
    // 4-lane group per segment: lane l -> seg = l>>2, sub = l&3.
    // 8 strided float4 loads cover the 128-float segment; cross-lane reduce
    // is only 2 shuffles (within the 4-lane group).
    {
        const int seg = lane >> 2;
        const int sub = lane & 3;
        for (int r = 0; r < ROWS_PER_WAVE; ++r) {
            const float* rp = x + (rowBase + r) * DCOLS + seg * SEG + sub * 4;
            float m = -__builtin_inff();
            #pragma unroll
            for (int i = 0; i < 8; ++i) {
                v4f v = *reinterpret_cast<const v4f*>(rp + i * 16);
                m = fmaxf(m, fmaxf(fmaxf(v.x, v.y), fmaxf(v.z, v.w)));
            }
            m = fmaxf(m, __shfl_xor(m, 1, 32));
            m = fmaxf(m, __shfl_xor(m, 2, 32));
            if (sub == 0) smax[wave * ROWS_PER_WAVE + r][seg] = m;
        }
    }
    __syncthreads();

    // ---------------- Phase 2: 16-row gate GEMM via V_WMMA_F32_16X16X4_F32 ---
    // A (16x4 f32 layout): lanes 0-15 hold M=lane, K={0,1}; lanes 16-31 hold
    // M=lane-16, K={2,3}. Two chained WMMAs cover K=0..7.
    {
        const int mrow  = lane & 15;
        const int kBase = (lane < 16) ? 0 : 2;
        const float* mx = &smax[wave * ROWS_PER_WAVE + mrow][0];

        v2f a0, a1;
        a0.x = mx[kBase + 0];
        a0.y = mx[kBase + 1];
        a1.x = mx[kBase + 4];
        a1.y = mx[kBase + 5];

        // B (4x16 f32 layout): lanes 0-15 hold N=lane, K={0,1}; lanes 16-31
        // hold N=lane-16, K={2,3}.  B[k][n] = W[n][k].  Columns N>=8 of D are
        // never read, so clamp the index (duplicate columns) instead of
        // branching — no exec-mask gymnastics around the loads.
        const int ncol = lane & 15;
        const int nc   = ncol & (FTN - 1);
        v2f b0, b1;
        b0.x = W[nc * FTN + kBase + 0];
        b0.y = W[nc * FTN + kBase + 1];
        b1.x = W[nc * FTN + kBase + 4];
        b1.y = W[nc * FTN + kBase + 5];

        v8f c = {};
        v8f d = __builtin_amdgcn_wmma_f32_16x16x4_f32(
                    false, a0, false, b0, (short)0, c, false, false);
        d = __builtin_amdgcn_wmma_f32_16x16x4_f32(
                    false, a1, false, b1, (short)0, d, false, false);

        // D layout: VGPR j, lanes 0-15 -> D[M=j][N=lane]; lanes 16-31 ->
        // D[M=j+8][N=lane-16]. Valid outputs at N < 8 (lanes 0-7, 16-23).
        if (ncol < FTN) {
            const float bn    = bias[ncol];
            const int   rbase = (lane < 16) ? 0 : 8;
            #pragma unroll
            for (int j = 0; j < 8; ++j) {
                const float t = d[j] + bn;
                const float e = __expf(-t);
                // fast sigmoid: single v_rcp_f32 instead of IEEE divide chain
                sscale[wave * ROWS_PER_WAVE + rbase + j][ncol] =
                    __builtin_amdgcn_rcpf(1.0f + e);
            }
        }
    }
    __syncthreads();

    // ---------------- Phase 3: scale + residual (re-read hits L2, NT store) --
    for (int r = 0; r < ROWS_PER_WAVE; ++r) {
        const float* rp = x   + (rowBase + r) * DCOLS + lane * 4;
        float*       op = out + (rowBase + r) * DCOLS + lane * 4;
        #pragma unroll
        for (int s = 0; s < FTN; ++s) {
            const float sc = 1.0f + sscale[wave * ROWS_PER_WAVE + r][s];
            v4f v = *reinterpret_cast<const v4f*>(rp + s * SEG);
            v4f o = v * sc;
            __builtin_nontemporal_store(o, reinterpret_cast<v4f*>(op + s * SEG));
        }
    }
}

extern "C" void kernel_launch(void* const* d_in, const int* in_sizes, int n_in,
                              void* d_out, int out_size, void* d_ws, size_t ws_size,
                              hipStream_t stream) {
    const float* x    = (const float*)d_in[0];
    const float* W    = (const float*)d_in[1];
    const float* bias = (const float*)d_in[2];
    float*       out  = (float*)d_out;

    const int rows   = in_sizes[0] / DCOLS;          // 65536
    const int blocks = rows / ROWS_PER_BLOCK;        // 512

    hipLaunchKernelGGL(importance_gate_kernel, dim3(blocks), dim3(256), 0, stream,
                       x, W, bias, out);
}